// HungarianAssigner3D_83021717831976
// MI455X (gfx1250) — compile-verified
//
#include <hip/hip_runtime.h>
#include <math.h>

// Problem constants (from reference setup_inputs)
#define QN   2048      // queries
#define GN   256       // ground truths
#define NC   10        // classes
#define DD   10        // box dims (first 8 used for L1)
#define NCP  12        // classes padded to multiple of K=4 for WMMA
#define ALPHA_C 0.25f
#define CLS_W   2.0f
#define REG_W   0.25f
#define FEPS    1e-12f

typedef __attribute__((ext_vector_type(2))) float v2f;
typedef __attribute__((ext_vector_type(8))) float v8f;

// ---------------------------------------------------------------------------
// Phase 1: per-(query,class) focal cost table, padded to 12 classes (zeros).
// Tiny: 2048*12 lanes, transcendental-bound, negligible.
// ---------------------------------------------------------------------------
__global__ void focal_table_kernel(const float* __restrict__ cls_preds,
                                   float* __restrict__ table) {
    int idx = blockIdx.x * blockDim.x + threadIdx.x;
    if (idx >= QN * NCP) return;
    int q = idx / NCP, c = idx % NCP;
    float out = 0.0f;
    if (c < NC) {
        float x   = cls_preds[q * NC + c];
        float p   = 1.0f / (1.0f + expf(-x));
        float omp = 1.0f - p;
        // GAMMA == 2 -> squares
        float pos = -logf(p   + FEPS) * ALPHA_C          * omp * omp;
        float neg = -logf(omp + FEPS) * (1.0f - ALPHA_C) * p   * p;
        out = (pos - neg) * CLS_W;
    }
    table[idx] = out;
}

// ---------------------------------------------------------------------------
// Phase 2: build transposed cost matrix aT[G][Q] (the small-side-rows layout
// the JV solver wants, since Q > G). Per 16x16 (g,q) tile:
//   C   = L1 cost (fused into the f32 accumulator)
//   D   = onehot(labels)[16x12] x focal_tableT[12x16]  via 3x WMMA f32 16x16x4
// One wave per tile; A/B/C/D laid out per the CDNA5 wave32 WMMA spec.
//   A 16x4 f32 : lane l -> M = l%16, VGPR v -> K = 2*(l/16)+v
//   B 4x16 f32 : lane l -> N = l%16, VGPR v -> K = 2*(l/16)+v
//   C/D 16x16  : lane l -> N = l%16, VGPR v -> M = v + 8*(l/16)
// ---------------------------------------------------------------------------
__global__ void cost_tiles_kernel(const int*   __restrict__ cls_gts,
                                  const float* __restrict__ table,
                                  const float* __restrict__ reg_gts,
                                  const float* __restrict__ reg_preds,
                                  float* __restrict__ aT) {
    const int lane = threadIdx.x & 31;
    const int wave = threadIdx.x >> 5;
    const int tile = blockIdx.x * (blockDim.x >> 5) + wave;
    const int qTiles = QN / 16;
    const int gt = tile / qTiles;
    const int qt = tile % qTiles;
    const int lm = lane & 15;
    const int lh = lane >> 4;
    const int q  = qt * 16 + lm;

    // L1 regression cost -> accumulator (C matrix)
    float rp[8];
#pragma unroll
    for (int d = 0; d < 8; ++d) rp[d] = reg_preds[q * DD + d];

    v8f acc;
#pragma unroll
    for (int vv = 0; vv < 8; ++vv) {
        const int g = gt * 16 + vv + 8 * lh;
        float s = 0.0f;
#pragma unroll
        for (int d = 0; d < 8; ++d) s += fabsf(rp[d] - reg_gts[g * DD + d]);
        acc[vv] = s * REG_W;
    }

    // One-hot label gather as dense GEMM: 3 chained WMMA f32 16x16x4
    const int lbl = cls_gts[gt * 16 + lm];
#pragma unroll
    for (int kb = 0; kb < NCP; kb += 4) {
        const int k0 = kb + 2 * lh;
        v2f a, b;
        a.x = (lbl == k0    ) ? 1.0f : 0.0f;
        a.y = (lbl == k0 + 1) ? 1.0f : 0.0f;
        b.x = table[q * NCP + k0];
        b.y = table[q * NCP + k0 + 1];
        acc = __builtin_amdgcn_wmma_f32_16x16x4_f32(
            /*neg_a=*/false, a, /*neg_b=*/false, b,
            /*c_mod=*/(short)0, acc, /*reuse_a=*/false, /*reuse_b=*/false);
    }

#pragma unroll
    for (int vv = 0; vv < 8; ++vv) {
        const int g = gt * 16 + vv + 8 * lh;
        aT[(size_t)g * QN + q] = acc[vv];
    }
}

// ---------------------------------------------------------------------------
// Phase 3: exact Jonker-Volgenant (e-maxx Hungarian) in ONE workgroup.
// All mutable state in LDS (~58 KB << 320 KB/WGP); only the 8 KB active cost
// row streams from L2 each inner iteration. Potentials in double to track the
// reference's float64 math; argmin ties broken by lowest index (np.argmin).
// 1024 threads = 32 wave32 waves, 2 columns per thread.
// ---------------------------------------------------------------------------
#define MM  QN       // large side (columns) = 2048
#define NN  GN       // small side (rows)    = 256
#define NTH 1024

__global__ __launch_bounds__(NTH)
void hungarian_kernel(const float* __restrict__ aT, float* __restrict__ out) {
    __shared__ double v_[MM + 1];
    __shared__ double minv_[MM + 1];
    __shared__ double u_[NN + 1];
    __shared__ int p_[MM + 1];
    __shared__ unsigned short way_[MM + 1];
    __shared__ unsigned char used_[MM + 1];
    __shared__ double rval[NTH];
    __shared__ unsigned short ridx[NTH];
    __shared__ int sj0, si0, sj1, sdone;
    __shared__ double sdelta;

    const int tid = threadIdx.x;
    for (int j = tid; j <= MM; j += NTH) { v_[j] = 0.0; p_[j] = 0; }
    if (tid <= NN) u_[tid] = 0.0;
    __syncthreads();

    for (int i = 1; i <= NN; ++i) {
        for (int j = tid; j <= MM; j += NTH) { minv_[j] = 1.0e300; used_[j] = 0; }
        if (tid == 0) { p_[0] = i; sj0 = 0; sdone = 0; }
        __syncthreads();

        while (true) {
            const int j0 = sj0;
            if (tid == 0) { used_[j0] = 1; si0 = p_[j0]; }
            __syncthreads();
            const int i0 = si0;
            const double ui0 = u_[i0];
            const float* arow = aT + (size_t)(i0 - 1) * MM;

            // scan free columns: relax minv/way, track local argmin
            double best = 1.0e301; int bidx = MM;
            for (int j = tid + 1; j <= MM; j += NTH) {
                if (!used_[j]) {
                    double cur = (double)arow[j - 1] - ui0 - v_[j];
                    if (cur < minv_[j]) { minv_[j] = cur; way_[j] = (unsigned short)j0; }
                    double mv = minv_[j];
                    if (mv < best || (mv == best && j < bidx)) { best = mv; bidx = j; }
                }
            }
            rval[tid] = best; ridx[tid] = (unsigned short)bidx;
            __syncthreads();
            // tree argmin, index tie-break == np.argmin semantics
            for (int s = NTH / 2; s > 0; s >>= 1) {
                if (tid < s) {
                    double ov = rval[tid + s]; int oi = ridx[tid + s];
                    if (ov < rval[tid] || (ov == rval[tid] && oi < (int)ridx[tid])) {
                        rval[tid] = ov; ridx[tid] = (unsigned short)oi;
                    }
                }
                __syncthreads();
            }
            if (tid == 0) { sj1 = ridx[0]; sdelta = minv_[sj1]; }
            __syncthreads();
            const int j1 = sj1; const double delta = sdelta;

            // potential updates (distinct p_[j] per used column -> race-free)
            for (int j = tid; j <= MM; j += NTH) {
                if (used_[j]) { u_[p_[j]] += delta; v_[j] -= delta; }
                else          { minv_[j] -= delta; }
            }
            if (tid == 0) { sj0 = j1; sdone = (p_[j1] == 0); }
            __syncthreads();
            if (sdone) break;
        }

        // augmenting-path flip (short sequential walk)
        if (tid == 0) {
            int j0c = sj0;
            while (j0c) { int j1c = way_[j0c]; p_[j0c] = p_[j1c]; j0c = j1c; }
        }
        __syncthreads();
    }

    // Extract matching. transposed=True in reference: r = column index (query),
    // c = assigned small-side row (gt). Ascending j == argsort(r) order.
    if (tid == 0) {
        int cnt = 0;
        for (int j = 1; j <= MM; ++j) {
            if (p_[j] > 0) {
                out[cnt]      = (float)(j - 1);      // r
                out[NN + cnt] = (float)(p_[j] - 1);  // c
                ++cnt;
            }
        }
    }
}

// ---------------------------------------------------------------------------
extern "C" void kernel_launch(void* const* d_in, const int* in_sizes, int n_in,
                              void* d_out, int out_size, void* d_ws, size_t ws_size,
                              hipStream_t stream) {
    const int*   cls_gts   = (const int*)  d_in[0];   // [G]
    const float* cls_preds = (const float*)d_in[1];   // [Q, NC]
    const float* reg_gts   = (const float*)d_in[2];   // [G, D]
    const float* reg_preds = (const float*)d_in[3];   // [Q, D]
    (void)d_in[4];                                    // box_weight unused
    (void)in_sizes; (void)n_in; (void)out_size; (void)ws_size;

    float* table = (float*)d_ws;                      // Q*NCP floats (~96 KB)
    float* aT    = table + (size_t)QN * NCP;          // G*Q floats (2 MB)

    {
        int total = QN * NCP;
        focal_table_kernel<<<(total + 255) / 256, 256, 0, stream>>>(cls_preds, table);
    }
    {
        int tiles = (GN / 16) * (QN / 16);            // 2048 tiles, 1 wave each
        cost_tiles_kernel<<<tiles / 8, 256, 0, stream>>>(cls_gts, table, reg_gts,
                                                         reg_preds, aT);
    }
    hungarian_kernel<<<1, NTH, 0, stream>>>(aT, (float*)d_out);
}